// RpnLoss_34677565948757
// MI455X (gfx1250) — compile-verified
//
#include <hip/hip_runtime.h>
#include <hip/hip_bf16.h>
#include <stdint.h>

// ---------------- problem constants (match reference) ----------------
#define A_TOTAL   1445      // NUM_ANCHOR(5) * FS(17) * FS(17)
#define HALF_A    723       // ceil(1445/2) for threefry odd-count padding
#define NB        256       // histogram buckets for O(A) top-k selection
#define BLK       256       // 8 wave32 per workgroup
#define POS_THR   0.6f
#define THR_LOW   0.3f
#define THR_HIGH  0.6f
#define N_POS_CAP 16
#define N_NEG_CAP 48

typedef unsigned int u32;
typedef u32   u32x4 __attribute__((ext_vector_type(4)));
typedef int   i32x4 __attribute__((ext_vector_type(4)));
typedef int   i32x8 __attribute__((ext_vector_type(8)));
typedef float v2f   __attribute__((ext_vector_type(2)));
typedef float v8f   __attribute__((ext_vector_type(8)));

// ---------------- CDNA5 feature detection (device pass only) ---------
#if defined(__has_builtin)
#  if __has_builtin(__builtin_amdgcn_tensor_load_to_lds)
#    define HAVE_TDM 1
#  endif
#  if __has_builtin(__builtin_amdgcn_wmma_f32_16x16x4_f32)
#    define HAVE_WMMA4 1
#  endif
#endif

// Diagnostic: surface which CDNA5 paths are active in compiler stderr.
#if defined(__AMDGCN__)
#  ifdef HAVE_TDM
#    warning "CDNA5 diag: TDM tensor_load_to_lds path ENABLED"
#  else
#    warning "CDNA5 diag: TDM builtin NOT FOUND - using fallback LDS copies"
#  endif
#  ifdef HAVE_WMMA4
#    warning "CDNA5 diag: WMMA f32 16x16x4 path ENABLED"
#  else
#    warning "CDNA5 diag: WMMA f32 16x16x4 builtin NOT FOUND - shuffle fallback"
#  endif
#endif

// ---------------- JAX threefry2x32 (exact) ---------------------------
__device__ __forceinline__ u32 rotl32(u32 x, int r) { return (x << r) | (x >> (32 - r)); }

__device__ __forceinline__ void threefry2x32(u32 k0, u32 k1, u32 x0, u32 x1,
                                             u32& o0, u32& o1) {
  const u32 k2 = 0x1BD11BDAu ^ k0 ^ k1;
  x0 += k0; x1 += k1;
#define TF_R(r) { x0 += x1; x1 = rotl32(x1, r); x1 ^= x0; }
  TF_R(13) TF_R(15) TF_R(26) TF_R(6)   x0 += k1; x1 += k2 + 1u;
  TF_R(17) TF_R(29) TF_R(16) TF_R(24)  x0 += k2; x1 += k0 + 2u;
  TF_R(13) TF_R(15) TF_R(26) TF_R(6)   x0 += k0; x1 += k1 + 3u;
  TF_R(17) TF_R(29) TF_R(16) TF_R(24)  x0 += k1; x1 += k2 + 4u;
  TF_R(13) TF_R(15) TF_R(26) TF_R(6)   x0 += k2; x1 += k0 + 5u;
#undef TF_R
  o0 = x0; o1 = x1;
}

__device__ __forceinline__ float bits_to_uniform(u32 b) {
  // JAX: bitcast((bits>>9)|0x3f800000) - 1.0  in [0,1)
  return __uint_as_float((b >> 9) | 0x3f800000u) - 1.0f;
}

__device__ __forceinline__ float softplus(float x) { // logaddexp(0,x), stable
  return fmaxf(x, 0.0f) + log1pf(expf(-fabsf(x)));
}

// ---------------- TDM: build D# and issue tensor_load_to_lds ---------
#ifdef HAVE_TDM
__device__ __forceinline__ void tdm_load_2d(u32 lds_byte_addr, const void* gptr,
                                            u32 tile_d0, u32 tile_d1, u32 stride_d0) {
  // D# group 0 (ISA 8.3): count=1 | lds_addr | global_addr[56:0] | type=2
  uint64_t ga = (uint64_t)(uintptr_t)gptr;
  u32x4 g0;
  g0[0] = 1u;
  g0[1] = lds_byte_addr;
  g0[2] = (u32)(ga & 0xffffffffu);
  g0[3] = (u32)((ga >> 32) & 0x01ffffffu) | 0x80000000u;
  // D# group 1 (ISA 8.4): mask=0, data_size=2(4B), dims/strides in elements.
  // tensor dims == tile dims (no OOB clipping needed); dim1_stride = dim0_stride.
  i32x8 g1;
  g1[0] = (int)0x00020000u;                                        // data_size=4B
  g1[1] = (int)((tile_d0 & 0xffffu) << 16);                        // tensor_dim0 lo16
  g1[2] = (int)(((tile_d0 >> 16) & 0xffffu) | ((tile_d1 & 0xffffu) << 16)); // d0 hi | d1 lo
  g1[3] = (int)(((tile_d1 >> 16) & 0xffffu) | ((tile_d0 & 0xffffu) << 16)); // d1 hi | tile_dim0
  g1[4] = (int)(tile_d1 & 0xffffu);                                // tile_dim1, tile_dim2=0
  g1[5] = (int)stride_d0;                                          // tensor_dim0_stride lo32
  g1[6] = (int)((stride_d0 & 0xffffu) << 16);  // d0_stride[47:32]=0 | d1_stride lo16
  g1[7] = 0;                                   // d1_stride[47:16] = 0
  i32x4 z4 = {0, 0, 0, 0};
#if __clang_major__ >= 23
  i32x8 z8 = {0, 0, 0, 0, 0, 0, 0, 0};
  __builtin_amdgcn_tensor_load_to_lds(g0, g1, z4, z4, z8, 0);
#else
  __builtin_amdgcn_tensor_load_to_lds(g0, g1, z4, z4, 0);
#endif
}

__device__ __forceinline__ void tdm_wait0() {
#if defined(__has_builtin) && __has_builtin(__builtin_amdgcn_s_wait_tensorcnt)
  __builtin_amdgcn_s_wait_tensorcnt(0);
#else
  asm volatile("s_wait_tensorcnt 0x0" ::: "memory");
#endif
}
#endif // HAVE_TDM

// ---------------- deterministic block reduction ----------------------
__device__ __forceinline__ float block_reduce(float v, float* buf, int tid) {
  buf[tid] = v;
  __syncthreads();
  for (int s = BLK / 2; s > 0; s >>= 1) {
    if (tid < s) buf[tid] += buf[tid + s];
    __syncthreads();
  }
  float r = buf[0];
  __syncthreads();
  return r;
}

// ---------------- main per-sample kernel -----------------------------
__global__ void __launch_bounds__(BLK)
rpn_main(const float* __restrict__ pred_cls, const float* __restrict__ pred_reg,
         const float* __restrict__ gt_bbox, const float* __restrict__ anc_center,
         const float* __restrict__ anc_corner, float* __restrict__ lc_out,
         float* __restrict__ lr_out, float* __restrict__ ac_out, int twoN) {
  __shared__ float s_cls[A_TOTAL];
  __shared__ float s_reg[4 * A_TOTAL];
  __shared__ float s_up[A_TOTAL];
  __shared__ float s_un[A_TOTAL];
  __shared__ unsigned char s_mask[A_TOTAL];
  __shared__ int s_hist_p[NB], s_hist_n[NB], s_pre_p[NB], s_pre_n[NB];
  __shared__ int s_npos, s_nneg;
  __shared__ float s_red[BLK];

  const int n = blockIdx.x;
  const int tid = threadIdx.x;

#ifdef HAVE_TDM
  // Wave 0 kicks the Tensor Data Mover: per-sample cls row (1445 f32) and
  // reg tile (4 x 1445 f32) stream into LDS while VALU does PRNG + IoU.
  if (tid < 32) {
    tdm_load_2d((u32)(uintptr_t)s_cls, pred_cls + (size_t)n * A_TOTAL,
                A_TOTAL, 1u, A_TOTAL);
    tdm_load_2d((u32)(uintptr_t)s_reg, pred_reg + (size_t)n * 4 * A_TOTAL,
                A_TOTAL, 4u, A_TOTAL);
  }
#endif

  // Per-sample keys = jax.random.split(key(42), 2N): kpos[n], kneg[n].
  u32 kpa, kpb, kna, knb;
  {
    u32 o0, o1;
    threefry2x32(0u, 42u, 2u * (u32)n,      (u32)twoN + 2u * (u32)n,      o0, o1);
    kpa = o0; kna = o1;
    threefry2x32(0u, 42u, 2u * (u32)n + 1u, (u32)twoN + 2u * (u32)n + 1u, o0, o1);
    kpb = o0; knb = o1;
  }

  if (tid == 0) { s_npos = 0; s_nneg = 0; }
  s_hist_p[tid] = 0;   // NB == BLK == 256
  s_hist_n[tid] = 0;
  __syncthreads();

  const float gx0 = gt_bbox[4 * n + 0], gy0 = gt_bbox[4 * n + 1];
  const float gx1 = gt_bbox[4 * n + 2], gy1 = gt_bbox[4 * n + 3];
  const float area_a = (gx1 - gx0) * (gy1 - gy0);

  int my_pos = 0, my_neg = 0;
  for (int a = tid; a < A_TOTAL; a += BLK) {
    float4 c = ((const float4*)anc_corner)[a];
    float iw = fmaxf(fminf(gx1, c.z) - fmaxf(gx0, c.x), 0.0f);
    float ih = fmaxf(fminf(gy1, c.w) - fmaxf(gy0, c.y), 0.0f);
    float inter = iw * ih;
    float area_b = (c.z - c.x) * (c.w - c.y);
    float iou = inter / (area_a + area_b - inter);
    int m = (iou > THR_HIGH) ? 1 : ((iou < THR_LOW) ? 0 : 2);
    s_mask[a] = (unsigned char)m;
    // uniform(key,(1445,)) = threefry over padded iota halves (odd count)
    u32 j  = (a < HALF_A) ? (u32)a : (u32)(a - HALF_A);
    u32 x1 = (j == (u32)(HALF_A - 1)) ? 0u : (u32)HALF_A + j;
    u32 o0, o1;
    threefry2x32(kpa, kpb, j, x1, o0, o1);
    float up = bits_to_uniform((a < HALF_A) ? o0 : o1);
    threefry2x32(kna, knb, j, x1, o0, o1);
    float un = bits_to_uniform((a < HALF_A) ? o0 : o1);
    s_up[a] = up;
    s_un[a] = un;
    if (m == 1) { my_pos++; atomicAdd(&s_hist_p[min((int)(up * NB), NB - 1)], 1); }
    else if (m == 0) { my_neg++; atomicAdd(&s_hist_n[min((int)(un * NB), NB - 1)], 1); }
  }
  if (my_pos) atomicAdd(&s_npos, my_pos);
  if (my_neg) atomicAdd(&s_nneg, my_neg);
  __syncthreads();

  // Exclusive prefix over buckets (<=255 LDS reads per thread — cheap).
  {
    int ap = 0, an = 0;
    for (int b = 0; b < tid; b++) { ap += s_hist_p[b]; an += s_hist_n[b]; }
    s_pre_p[tid] = ap;
    s_pre_n[tid] = an;
  }

#ifdef HAVE_TDM
  if (tid < 32) tdm_wait0();   // issuing wave drains TENSORcnt, barrier publishes
#else
  for (int i = tid; i < A_TOTAL; i += BLK)
    s_cls[i] = pred_cls[(size_t)n * A_TOTAL + i];
  for (int i = tid; i < 4 * A_TOTAL; i += BLK)
    s_reg[i] = pred_reg[(size_t)n * 4 * A_TOTAL + i];
#endif
  __syncthreads();

  const int n_pos = s_npos, n_neg = s_nneg;
  const int k_p = min(n_pos, N_POS_CAP);
  const int k_n = min(n_neg, (n_pos > 0) ? 3 * n_pos : N_NEG_CAP);
  const float denom = (float)max(k_p + k_n, 1);

  const float gcx = (gx0 + gx1) * 0.5f, gcy = (gy0 + gy1) * 0.5f;
  const float gw = gx1 - gx0, gh = gy1 - gy0;

  float lc = 0.0f, lr = 0.0f, accc = 0.0f;
  for (int a = tid; a < A_TOTAL; a += BLK) {
    int m = s_mask[a];
    float c = s_cls[a];
    bool pos = (m == 1);
    accc += ((c >= POS_THR) == pos) ? 1.0f : 0.0f;   // acc: (cls>=thr)==pos
    if (pos) {
      // top-k_p smallest uniforms among positives (stable-argsort semantics)
      float u = s_up[a];
      int b = min((int)(u * NB), NB - 1);
      int base = s_pre_p[b];
      bool sel;
      if (base >= k_p) sel = false;
      else if (base + s_hist_p[b] <= k_p) sel = true;
      else { // boundary bucket: exact in-bucket rank with index tie-break
        int r = 0;
        for (int q = 0; q < A_TOTAL; q++) {
          if (s_mask[q] == 1) {
            float uq = s_up[q];
            if (min((int)(uq * NB), NB - 1) == b &&
                (uq < u || (uq == u && q < a))) r++;
          }
        }
        sel = (base + r) < k_p;
      }
      if (sel) lc += softplus(-c);
      float4 ctr = ((const float4*)anc_center)[a];     // (cx, cy, w, h)
      float t0 = (gcx - ctr.x) / ctr.z;
      float t1 = (gcy - ctr.y) / ctr.w;
      float t2 = logf(gw / ctr.z);
      float t3 = logf(gh / ctr.w);
      float d0 = s_reg[0 * A_TOTAL + a] - t0;
      float d1 = s_reg[1 * A_TOTAL + a] - t1;
      float d2 = s_reg[2 * A_TOTAL + a] - t2;
      float d3 = s_reg[3 * A_TOTAL + a] - t3;
      lr += d0 * d0 + d1 * d1 + d2 * d2 + d3 * d3;
    } else if (m == 0) {
      float u = s_un[a];
      int b = min((int)(u * NB), NB - 1);
      int base = s_pre_n[b];
      bool sel;
      if (base >= k_n) sel = false;
      else if (base + s_hist_n[b] <= k_n) sel = true;
      else {
        int r = 0;
        for (int q = 0; q < A_TOTAL; q++) {
          if (s_mask[q] == 0) {
            float uq = s_un[q];
            if (min((int)(uq * NB), NB - 1) == b &&
                (uq < u || (uq == u && q < a))) r++;
          }
        }
        sel = (base + r) < k_n;
      }
      if (sel) lc += softplus(c);
    }
  }

  lc = block_reduce(lc, s_red, tid);
  lr = block_reduce(lr, s_red, tid);
  accc = block_reduce(accc, s_red, tid);
  if (tid == 0) {
    lc_out[n] = lc / denom;
    lr_out[n] = (n_pos > 0) ? lr / (4.0f * (float)n_pos) : 0.0f;
    ac_out[n] = accc / (float)A_TOTAL;
  }
}

// ---------------- final batch reduction (one wave32) ------------------
// Uses v_wmma_f32_16x16x4_f32 with an all-ones B (layout-independent) as a
// deterministic fixed-order reduction: D[m][*] accumulates row sums of A,
// 64 elements consumed per WMMA; finish with 8 C-VGPR adds + shfl_xor(16).
__global__ void __launch_bounds__(32)
rpn_reduce(const float* __restrict__ lc, const float* __restrict__ lr,
           const float* __restrict__ ac, float* __restrict__ out, int n) {
  const int l = threadIdx.x;
  const float* arrs[3] = {lc, lr, ac};
  float sums[3];
#ifdef HAVE_WMMA4
  v2f bones;
  bones[0] = 1.0f;
  bones[1] = 1.0f;
  for (int q = 0; q < 3; q++) {
    v8f cacc;
    for (int i = 0; i < 8; i++) cacc[i] = 0.0f;
    const float* x = arrs[q];
    for (int base = 0; base + 64 <= n; base += 64) {
      v2f a;
      a[0] = x[base + l];
      a[1] = x[base + 32 + l];
      cacc = __builtin_amdgcn_wmma_f32_16x16x4_f32(
          false, a, false, bones, (short)0, cacc, false, false);
    }
    float s = cacc[0] + cacc[1] + cacc[2] + cacc[3] +
              cacc[4] + cacc[5] + cacc[6] + cacc[7];   // half-sum per lane-half
    s += __shfl_xor(s, 16, 32);                        // full 64-per-iter sum
    sums[q] = s;
  }
#else
  for (int q = 0; q < 3; q++) {
    const float* x = arrs[q];
    float s = 0.0f;
    for (int i = l; i < n; i += 32) s += x[i];
    for (int o = 16; o > 0; o >>= 1) s += __shfl_xor(s, o, 32);
    sums[q] = s;
  }
#endif
  if (l == 0) {
    float inv = 1.0f / (float)n;
    float mc = sums[0] * inv, mr = sums[1] * inv, ma = sums[2] * inv;
    out[0] = mc + mr;  // loss_total (CLS_W = REG_W = 1)
    out[1] = mc;       // loss_cls
    out[2] = mr;       // loss_reg
    out[3] = ma;       // acc_cls
  }
}

// ---------------- entry point ----------------------------------------
extern "C" void kernel_launch(void* const* d_in, const int* in_sizes, int n_in,
                              void* d_out, int out_size, void* d_ws, size_t ws_size,
                              hipStream_t stream) {
  (void)n_in; (void)out_size; (void)ws_size;
  const float* pred_cls   = (const float*)d_in[0];
  const float* pred_reg   = (const float*)d_in[1];
  const float* gt_bbox    = (const float*)d_in[2];
  const float* anc_center = (const float*)d_in[3];
  const float* anc_corner = (const float*)d_in[4];
  const int N = in_sizes[0] / A_TOTAL;   // 4096

  float* lc = (float*)d_ws;
  float* lr = lc + N;
  float* ac = lr + N;

  rpn_main<<<N, BLK, 0, stream>>>(pred_cls, pred_reg, gt_bbox, anc_center,
                                  anc_corner, lc, lr, ac, 2 * N);
  rpn_reduce<<<1, 32, 0, stream>>>(lc, lr, ac, (float*)d_out, N);
}